// _FeaturePropagation_86165633892449
// MI455X (gfx1250) — compile-verified
//
#include <hip/hip_runtime.h>
#include <hip/hip_bf16.h>

typedef __attribute__((ext_vector_type(2))) float v2f;
typedef __attribute__((ext_vector_type(8))) float v8f;

#define B_  8
#define N_  8192
#define S_  2048
#define C1_ 128
#define C2_ 256
#define KCAT_ 384   // C1 + C2
#define M_  (B_ * N_)   // 65536 rows

// ---------------------------------------------------------------------------
// Kernel 1: 3-NN search over xyz2 (staged fully in LDS: 2048*3*4 = 24 KB),
// writes per-point neighbor indices and normalized inverse-distance weights.
// Grid: B_ * (N_/256) blocks of 256 threads; each thread owns one query point.
// ---------------------------------------------------------------------------
__global__ void knn_kernel(const float* __restrict__ xyz1,
                           const float* __restrict__ xyz2,
                           int* __restrict__ idx_out,
                           float* __restrict__ wgt_out) {
    __shared__ float sx[S_], sy[S_], sz[S_];
    const int b    = blockIdx.x / (N_ / 256);
    const int nblk = blockIdx.x % (N_ / 256);
    const int tid  = threadIdx.x;

    const float* src = xyz2 + (size_t)b * S_ * 3;
    for (int s = tid; s < S_; s += 256) {
        sx[s] = src[s * 3 + 0];
        sy[s] = src[s * 3 + 1];
        sz[s] = src[s * 3 + 2];
    }
    __syncthreads();

    const int n = nblk * 256 + tid;
    const float* q = xyz1 + ((size_t)b * N_ + n) * 3;
    const float qx = q[0], qy = q[1], qz = q[2];

    float d0 = 3.4e38f, d1 = 3.4e38f, d2 = 3.4e38f;
    int   i0 = 0,       i1 = 0,       i2 = 0;

    for (int s = 0; s < S_; ++s) {
        const float dx = qx - sx[s];
        const float dy = qy - sy[s];
        const float dz = qz - sz[s];
        const float d  = dx * dx + dy * dy + dz * dz;
        // strict < keeps the earliest index on ties, matching lax.top_k
        if (d < d0)      { d2 = d1; i2 = i1; d1 = d0; i1 = i0; d0 = d; i0 = s; }
        else if (d < d1) { d2 = d1; i2 = i1; d1 = d;  i1 = s; }
        else if (d < d2) { d2 = d;  i2 = s; }
    }

    const float r0 = 1.0f / (d0 + 1e-8f);
    const float r1 = 1.0f / (d1 + 1e-8f);
    const float r2 = 1.0f / (d2 + 1e-8f);
    const float inv = 1.0f / (r0 + r1 + r2);

    const size_t base = ((size_t)b * N_ + n) * 3;
    idx_out[base + 0] = i0;  wgt_out[base + 0] = r0 * inv;
    idx_out[base + 1] = i1;  wgt_out[base + 1] = r1 * inv;
    idx_out[base + 2] = i2;  wgt_out[base + 2] = r2 * inv;
}

// ---------------------------------------------------------------------------
// Kernel 2: weighted 3-neighbor interpolation of points2 + concat of points1.
// One block (256 threads) per query point; thread = channel -> coalesced
// gathers along the C2=256 channel axis of points2.
// Output row layout: [ points1 (128) | interp (256) ]  (KCAT_ = 384)
// ---------------------------------------------------------------------------
__global__ void interp_concat_kernel(const float* __restrict__ points1,
                                     const float* __restrict__ points2,
                                     const int* __restrict__ idx_in,
                                     const float* __restrict__ wgt_in,
                                     float* __restrict__ xcat) {
    const int p = blockIdx.x;            // 0 .. M_-1
    const int b = p / N_;
    const int c = threadIdx.x;           // 0 .. 255

    const size_t base = (size_t)p * 3;
    const int   i0 = idx_in[base + 0], i1 = idx_in[base + 1], i2 = idx_in[base + 2];
    const float w0 = wgt_in[base + 0], w1 = wgt_in[base + 1], w2 = wgt_in[base + 2];

    const float* p2 = points2 + (size_t)b * S_ * C2_;
    const float v = w0 * p2[(size_t)i0 * C2_ + c]
                  + w1 * p2[(size_t)i1 * C2_ + c]
                  + w2 * p2[(size_t)i2 * C2_ + c];

    float* row = xcat + (size_t)p * KCAT_;
    row[C1_ + c] = v;
    if (c < C1_) row[c] = points1[(size_t)p * C1_ + c];
}

// ---------------------------------------------------------------------------
// Kernel 3: register-blocked GEMM  C[M,Nc] = A[M,K] * W[Nc,K]^T + bias (fp32)
// One wave32 computes a 64x64 output block = 4x4 grid of 16x16 WMMA tiles
// (V_WMMA_F32_16X16X4_F32). K and Nc are template constants so every
// fragment address is (one base VGPR pair) + 24-bit immediate offset:
// A frag i lives at abase + i*16*K + k, W frag j at wbase + j*16*K + k.
// Per k-step: 8x global_load_b64 feed 16 WMMAs (4x fragment reuse).
// __launch_bounds__(256,1) = min 1 wave/EU: this wave needs ~160 VGPRs
// (128 accumulator + operands); low occupancy is fine, spilling is not.
// ISA f32 operand layout: lanes 0-15 carry K=0,1 in 2 VGPRs (A: M=lane%16,
// B: N=lane%16), lanes 16-31 carry K=2,3. D: VGPR r = row 8*(lane/16)+r,
// col = lane%16. M,Nc multiples of 64 -> no tails, EXEC all-1s at WMMAs.
// ---------------------------------------------------------------------------
template <int K, int NC>
__global__ __launch_bounds__(256, 1)
void gemm_wmma64_kernel(const float* __restrict__ A,
                        const float* __restrict__ W,
                        const float* __restrict__ bias,
                        float* __restrict__ C,
                        int M) {
    const int wave = (blockIdx.x * blockDim.x + threadIdx.x) >> 5;
    const int lane = threadIdx.x & 31;
    const int nb64 = NC >> 6;            // 64-col blocks
    const int wm = wave / nb64;
    const int wn = wave % nb64;
    if (wm * 64 >= M) return;            // wave-uniform; EXEC stays all-1s

    const int half = lane >> 4;          // 0: K=0,1   1: K=2,3
    const int l    = lane & 15;

    const float* abase = A + (size_t)(wm * 64 + l) * K + half * 2;
    const float* wbase = W + (size_t)(wn * 64 + l) * K + half * 2;

    v8f acc[4][4];
    #pragma unroll
    for (int i = 0; i < 4; ++i)
        #pragma unroll
        for (int j = 0; j < 4; ++j)
            acc[i][j] = (v8f){0.f, 0.f, 0.f, 0.f, 0.f, 0.f, 0.f, 0.f};

    #pragma unroll 2
    for (int k = 0; k < K; k += 4) {
        v2f af[4], bf[4];
        #pragma unroll
        for (int i = 0; i < 4; ++i)
            af[i] = *(const v2f*)(abase + i * 16 * K + k);   // imm offset
        #pragma unroll
        for (int j = 0; j < 4; ++j)
            bf[j] = *(const v2f*)(wbase + j * 16 * K + k);   // imm offset
        #pragma unroll
        for (int i = 0; i < 4; ++i)
            #pragma unroll
            for (int j = 0; j < 4; ++j)
                acc[i][j] = __builtin_amdgcn_wmma_f32_16x16x4_f32(
                    /*neg_a=*/false, af[i], /*neg_b=*/false, bf[j],
                    /*c_mod=*/(short)0, acc[i][j],
                    /*reuse_a=*/false, /*reuse_b=*/false);
    }

    const float bv[4] = { bias[wn * 64 + 0 * 16 + l], bias[wn * 64 + 1 * 16 + l],
                          bias[wn * 64 + 2 * 16 + l], bias[wn * 64 + 3 * 16 + l] };
    float* cbase = C + (size_t)(wm * 64 + half * 8) * NC + wn * 64 + l;
    #pragma unroll
    for (int i = 0; i < 4; ++i)
        #pragma unroll
        for (int r = 0; r < 8; ++r)
            #pragma unroll
            for (int j = 0; j < 4; ++j)
                cbase[(size_t)(i * 16 + r) * NC + j * 16] = acc[i][j][r] + bv[j];
}

// ---------------------------------------------------------------------------
// Kernel 4: per-channel sum / sum-of-squares for BatchNorm batch statistics.
// Thread owns channel c = threadIdx.x -> fully coalesced row-major reads;
// partials merged with f32 global atomics (accumulators memset to 0 first).
// ---------------------------------------------------------------------------
__global__ void colstats_kernel(const float* __restrict__ Y, int M, int Nc,
                                float* __restrict__ sum, float* __restrict__ sumsq) {
    const int c = threadIdx.x;           // blockDim.x == Nc
    const int rowsPer = M / gridDim.x;
    const int r0 = blockIdx.x * rowsPer;
    float s = 0.f, s2 = 0.f;
    for (int r = r0; r < r0 + rowsPer; ++r) {
        const float v = Y[(size_t)r * Nc + c];
        s += v;
        s2 += v * v;
    }
    atomicAdd(&sum[c], s);
    atomicAdd(&sumsq[c], s2);
}

// ---------------------------------------------------------------------------
// Kernel 5: BatchNorm (population var = E[x^2]-E[x]^2, matching jnp.var) +
// affine + ReLU, elementwise.
// ---------------------------------------------------------------------------
__global__ void bnrelu_kernel(const float* __restrict__ Y,
                              const float* __restrict__ sum,
                              const float* __restrict__ sumsq,
                              const float* __restrict__ gamma,
                              const float* __restrict__ beta,
                              int M, int Nc, float* __restrict__ out) {
    const size_t i = (size_t)blockIdx.x * blockDim.x + threadIdx.x;
    const int c = (int)(i % Nc);
    const float invM = 1.0f / (float)M;
    const float mean = sum[c] * invM;
    const float var  = sumsq[c] * invM - mean * mean;
    const float v = (Y[i] - mean) * rsqrtf(var + 1e-5f) * gamma[c] + beta[c];
    out[i] = v > 0.f ? v : 0.f;
}

// ---------------------------------------------------------------------------
// Launch
// ---------------------------------------------------------------------------
extern "C" void kernel_launch(void* const* d_in, const int* in_sizes, int n_in,
                              void* d_out, int out_size, void* d_ws, size_t ws_size,
                              hipStream_t stream) {
    const float* xyz1    = (const float*)d_in[0];
    const float* xyz2    = (const float*)d_in[1];
    const float* points1 = (const float*)d_in[2];
    const float* points2 = (const float*)d_in[3];
    const float* w0      = (const float*)d_in[4];
    const float* b0      = (const float*)d_in[5];
    const float* g0      = (const float*)d_in[6];
    const float* be0     = (const float*)d_in[7];
    const float* w1      = (const float*)d_in[8];
    const float* b1      = (const float*)d_in[9];
    const float* g1      = (const float*)d_in[10];
    const float* be1     = (const float*)d_in[11];
    float* out = (float*)d_out;

    // Workspace layout (bytes, 256-aligned blocks):
    char* ws = (char*)d_ws;
    float* stats = (float*)ws;                       // 1024 floats: sum1|sq1|sum2|sq2
    float* sum1  = stats + 0;
    float* sq1   = stats + 256;
    float* sum2  = stats + 512;
    float* sq2   = stats + 640;
    int*   idx   = (int*)  (ws + 4096);                      // M*3 ints   (768 KB)
    float* wgt   = (float*)(ws + 4096 + 786432);             // M*3 f32    (768 KB)
    float* xcat  = (float*)(ws + 1576960);                   // M*384 f32  (96 MB)
    float* y1    = (float*)(ws + 102240256);                 // M*256 f32  (64 MB)
    float* y2    = (float*)(ws + 169349120);                 // M*128 f32  (32 MB)

    hipMemsetAsync(stats, 0, 4096, stream);

    // 1) 3-NN
    knn_kernel<<<B_ * (N_ / 256), 256, 0, stream>>>(xyz1, xyz2, idx, wgt);

    // 2) interpolate + concat
    interp_concat_kernel<<<M_, 256, 0, stream>>>(points1, points2, idx, wgt, xcat);

    // 3) GEMM1: [M,384] x [256,384]^T -> y1, one wave per 64x64 block
    {
        const int waves = (M_ / 64) * (C2_ / 64);    // 4096 waves
        gemm_wmma64_kernel<KCAT_, C2_><<<waves * 32 / 256, 256, 0, stream>>>(
            xcat, w0, b0, y1, M_);
    }
    // 4) BN stats + BN/ReLU (in place on y1)
    colstats_kernel<<<256, C2_, 0, stream>>>(y1, M_, C2_, sum1, sq1);
    bnrelu_kernel<<<(M_ * C2_) / 256, 256, 0, stream>>>(y1, sum1, sq1, g0, be0, M_, C2_, y1);

    // 5) GEMM2: [M,256] x [128,256]^T -> y2, one wave per 64x64 block
    {
        const int waves = (M_ / 64) * (C1_ / 64);    // 2048 waves
        gemm_wmma64_kernel<C2_, C1_><<<waves * 32 / 256, 256, 0, stream>>>(
            y1, w1, b1, y2, M_);
    }
    // 6) BN stats + BN/ReLU -> d_out
    colstats_kernel<<<256, C1_, 0, stream>>>(y2, M_, C1_, sum2, sq2);
    bnrelu_kernel<<<(M_ * C1_) / 256, 256, 0, stream>>>(y2, sum2, sq2, g1, be1, M_, C1_, out);
}